// MicroCommunity_79010218377218
// MI455X (gfx1250) — compile-verified
//
#include <hip/hip_runtime.h>

// ---------------------------------------------------------------------------
// MicroCommunity kernel for MI455X (gfx1250, wave32)
// Memory-bound workload (~150MB traffic, ~100 MFLOP). Uses
// V_WMMA_F32_16X16X4_F32 for the only dense matmul (h @ W2^T) over 16-row
// tiles, LDS for the tile transpose, float4 (b128) streaming for the einsum,
// hardware v_rcp_f32 for sigmoid/normalization, wave32 shfl-xor loss
// reduction, and global f32 atomics for the scatter-add.
// ---------------------------------------------------------------------------

typedef float v2f __attribute__((ext_vector_type(2)));
typedef float v8f __attribute__((ext_vector_type(8)));

#define C_CLASSES 10000
#define M_CLASS 6
#define D_DIM 256
#define BS 16384
#define EPSV 1e-4f

#define WAVES_PER_BLOCK 8
#define ROWS_PER_WAVE 16
#define ROWS_PER_BLOCK (WAVES_PER_BLOCK * ROWS_PER_WAVE) /* 128 */
#define THREADS (WAVES_PER_BLOCK * 32)                   /* 256 */

// Output layout in d_out (floats):
//   [0]                      loss
//   [1 .. 1+BS*M)            sum_v   (16384 x 6)
//   [.. + C*M*D)             new_memory (60000 x 256)
//   [.. + C*M)               new_memory_weights (60000)
#define SUMV_N ((long)BS * M_CLASS)
#define MEM_N ((long)C_CLASSES * M_CLASS * D_DIM)
#define MW_N ((long)C_CLASSES * M_CLASS)

__global__ void mc_init(const float* __restrict__ memory,
                        const float* __restrict__ memw,
                        float* __restrict__ out) {
  long idx = (long)blockIdx.x * blockDim.x + threadIdx.x;
  if (idx < MEM_N) {
    out[1 + SUMV_N + idx] = memory[idx];
  } else if (idx < MEM_N + MW_N) {
    out[1 + SUMV_N + MEM_N + (idx - MEM_N)] = memw[idx - MEM_N];
  } else if (idx == MEM_N + MW_N) {
    out[0] = 0.0f; // loss accumulator
  }
}

__global__ __launch_bounds__(THREADS) void mc_main(
    const float* __restrict__ data, const int* __restrict__ labels,
    const float* __restrict__ beta, const float* __restrict__ centers,
    const float* __restrict__ W1, const float* __restrict__ b1,
    const float* __restrict__ W2, const float* __restrict__ b2,
    float* __restrict__ out) {
  __shared__ float s_pre[WAVES_PER_BLOCK][16][16];      // WMMA D tile transpose
  __shared__ float s_nw[WAVES_PER_BLOCK][16][M_CLASS];  // per-row norm_w
  __shared__ int s_lab[WAVES_PER_BLOCK][16];            // per-row label

  const int tid = threadIdx.x;
  const int wave = tid >> 5;
  const int lane = tid & 31;
  const int half = lane >> 4;  // 0: lanes 0-15, 1: lanes 16-31
  const int n = lane & 15;     // tile row (for A) / tile col (for B,D)

  const int rowBase = blockIdx.x * ROWS_PER_BLOCK + wave * ROWS_PER_WAVE;
  const int myRow = rowBase + n;  // duplicated across the two half-waves

  // ---- Phase 1: tiny MLP front-end, per 16-row tile via WMMA -------------
  // h[m] = relu(W1[m, label] + b1[m])
  const int lab = labels[myRow];
  float h[M_CLASS];
#pragma unroll
  for (int m = 0; m < M_CLASS; ++m) {
    float v = W1[(long)m * C_CLASSES + lab] + b1[m];
    h[m] = v > 0.0f ? v : 0.0f;
  }

  // A (16xK f32 layout): lanes 0-15 hold K=0 (v0) and K=1 (v1) of row n;
  // lanes 16-31 hold K=2 (v0) and K=3 (v1). Second WMMA covers K=4..7
  // (K=6,7 zero padding).
  v2f a0, a1;
  a0.x = half ? h[2] : h[0];
  a0.y = half ? h[3] : h[1];
  a1.x = half ? 0.0f : h[4];
  a1.y = half ? 0.0f : h[5];

  // B (Kx16 f32 layout, mirrored): B[k][n] = W2^T[k][n] = W2[n][k], n<6.
  const bool nvalid = (n < M_CLASS);
  v2f bm0, bm1;
  bm0.x = nvalid ? W2[n * M_CLASS + (half ? 2 : 0)] : 0.0f;
  bm0.y = nvalid ? W2[n * M_CLASS + (half ? 3 : 1)] : 0.0f;
  bm1.x = (nvalid && !half) ? W2[n * M_CLASS + 4] : 0.0f;
  bm1.y = (nvalid && !half) ? W2[n * M_CLASS + 5] : 0.0f;

  // C preloaded with bias b2[n] (every row of the 16x16 tile sees b2[N]).
  float bias = nvalid ? b2[n] : 0.0f;
  v8f acc;
#pragma unroll
  for (int j = 0; j < 8; ++j) acc[j] = bias;

  // D = A*B + C  (pre-sigmoid activations for 16 rows x 6 units)
  acc = __builtin_amdgcn_wmma_f32_16x16x4_f32(false, a0, false, bm0, (short)0,
                                              acc, false, false);
  acc = __builtin_amdgcn_wmma_f32_16x16x4_f32(false, a1, false, bm1, (short)0,
                                              acc, false, false);

  // D layout: VGPR j -> (M=j, N=lane) for lanes 0-15, (M=j+8, N=lane-16) for
  // lanes 16-31. Transpose through LDS so each lane can read its row.
#pragma unroll
  for (int j = 0; j < 8; ++j) {
    s_pre[wave][j + (half ? 8 : 0)][n] = acc[j];
  }
  __syncthreads();

  // ---- Phase 2: per-row epilogue (sigmoid, cumsum, weights) --------------
  float* sumv_out = out + 1;
  float* mem_out = out + 1 + SUMV_N;
  float* memw_out = mem_out + MEM_N;

  {
    const float bet = beta[myRow];
    float cs[M_CLASS], ww[M_CLASS];
    float run = 0.0f, wsum = 0.0f;
#pragma unroll
    for (int m = 0; m < M_CLASS; ++m) {
      float pre = s_pre[wave][n][m];
      // sigmoid via hardware v_rcp_f32 (avoids IEEE div_scale chains)
      float o = __builtin_amdgcn_rcpf(1.0f + __expf(-pre)) + EPSV;
      run += o;
      cs[m] = run;
      float d = bet - run;
      float wv = __expf(-sqrtf(d * d + 1e-10f));
      ww[m] = wv;
      wsum += wv;
    }
    float inv = __builtin_amdgcn_rcpf(wsum + EPSV + 1e-10f);
    if (half == 0) {  // one writer per row
      s_lab[wave][n] = lab;
#pragma unroll
      for (int m = 0; m < M_CLASS; ++m) {
        float nw = ww[m] * inv;
        s_nw[wave][n][m] = nw;
        sumv_out[(long)myRow * M_CLASS + m] = cs[m];
        atomicAdd(&memw_out[(long)lab * M_CLASS + m], nw);
      }
    }
  }
  __syncthreads();

  // ---- Phase 3: streaming einsum + loss + scatter-add --------------------
  // Wave processes its 16 rows; 32 lanes x 8 floats (2x b128) cover D=256.
  float lossAcc = 0.0f;
  const int d0 = lane * 8;

  for (int i = 0; i < ROWS_PER_WAVE; ++i) {
    const int r = rowBase + i;
    const int lr = s_lab[wave][i];
    float nw[M_CLASS];
#pragma unroll
    for (int m = 0; m < M_CLASS; ++m) nw[m] = s_nw[wave][i][m];

    if (i + 1 < ROWS_PER_WAVE) {  // prefetch next row's streams (L2-resident)
      __builtin_prefetch(data + (long)(r + 1) * D_DIM + d0, 0, 1);
      __builtin_prefetch(
          centers + ((long)s_lab[wave][i + 1] * M_CLASS) * D_DIM + d0, 0, 1);
    }

    const float* drow = data + (long)r * D_DIM + d0;
    const float4 x0 = *(const float4*)(drow);
    const float4 x1 = *(const float4*)(drow + 4);
    float x[8] = {x0.x, x0.y, x0.z, x0.w, x1.x, x1.y, x1.z, x1.w};

    float cm[8];
#pragma unroll
    for (int e = 0; e < 8; ++e) cm[e] = 0.0f;

#pragma unroll
    for (int m = 0; m < M_CLASS; ++m) {
      const long crBase = ((long)lr * M_CLASS + m) * D_DIM + d0;
      const float* crow = centers + crBase;
      float* mrow = mem_out + crBase;
      const float4 g0 = *(const float4*)(crow);
      const float4 g1 = *(const float4*)(crow + 4);
      const float g[8] = {g0.x, g0.y, g0.z, g0.w, g1.x, g1.y, g1.z, g1.w};
      const float w = nw[m];
#pragma unroll
      for (int e = 0; e < 8; ++e) {
        cm[e] += w * g[e];
        atomicAdd(&mrow[e], w * x[e]);  // feat_cent scatter-add
      }
    }

#pragma unroll
    for (int e = 0; e < 8; ++e) {
      float diff = x[e] - cm[e];
      lossAcc += diff * diff;
    }
  }

  // Wave32 butterfly reduction -> one global atomic per wave.
  lossAcc *= 1.0f / ((float)BS * (float)D_DIM);
#pragma unroll
  for (int off = 16; off > 0; off >>= 1) {
    lossAcc += __shfl_xor(lossAcc, off, 32);
  }
  if (lane == 0) atomicAdd(out, lossAcc);
}

extern "C" void kernel_launch(void* const* d_in, const int* in_sizes, int n_in,
                              void* d_out, int out_size, void* d_ws,
                              size_t ws_size, hipStream_t stream) {
  const float* data = (const float*)d_in[0];
  const int* labels = (const int*)d_in[1];
  const float* beta = (const float*)d_in[2];
  const float* centers = (const float*)d_in[3];
  const float* W1 = (const float*)d_in[4];
  const float* b1 = (const float*)d_in[5];
  const float* W2 = (const float*)d_in[6];
  const float* b2 = (const float*)d_in[7];
  const float* memory = (const float*)d_in[8];
  const float* memw = (const float*)d_in[9];
  float* out = (float*)d_out;

  const long initTotal = MEM_N + MW_N + 1;
  const int initBlocks = (int)((initTotal + 255) / 256);
  mc_init<<<initBlocks, 256, 0, stream>>>(memory, memw, out);

  const int mainBlocks = BS / ROWS_PER_BLOCK;  // 128
  mc_main<<<mainBlocks, THREADS, 0, stream>>>(data, labels, beta, centers, W1,
                                              b1, W2, b2, out);
}